// Phi_Layer_32212254720394
// MI455X (gfx1250) — compile-verified
//
#include <hip/hip_runtime.h>

typedef __attribute__((ext_vector_type(2))) float v2f;
typedef __attribute__((ext_vector_type(8))) float v8f;

#define B_  16
#define NT  64
#define NR  32
#define MT  32
#define MR  16
#define G2_ 4096
#define GT  16
#define KC  8            // k's per chunk (one per wave)
#define NCH (NT / KC)    // 8 chunks

static __device__ __forceinline__ v8f wmma_f32(v2f a, v2f b, v8f c) {
  // (neg_a, A, neg_b, B, c_mod, C, reuse_a, reuse_b)
  return __builtin_amdgcn_wmma_f32_16x16x4_f32(false, a, false, b, (short)0, c,
                                               false, false);
}

__global__ __launch_bounds__(256)
void phi_kernel(const float* __restrict__ W,   // [B, Nr, Mr, 2]
                const float* __restrict__ F,   // [B, Nt, Mt, 2]
                const float* __restrict__ K2,  // [Nt*Nr, G2, 2]
                float* __restrict__ out) {     // [B, Mt*Mr, G2, 2]
  const int g0 = blockIdx.x * GT;
  const int b  = blockIdx.y;

  const int tid  = threadIdx.x;
  const int wave = tid >> 5;
  const int lane = tid & 31;
  const int half = lane >> 4;
  const int l16  = lane & 15;

  // double-buffered stage-1 staging: [buf][(kk*2+part)][row j][col gg]
  __shared__ float tmpL[2][KC * 2 * 16 * 16];   // 32 KB

  // ---- Stage-1 A-sets (loop invariant), Gauss 3-mult form ----
  // Bm[j,l] = conj(W[b,l,j]) = Wre - i*Wim
  // tRe = Wre@Kr + Wim@Ki           = P1 + P2
  // tIm = Wre@Ki - Wim@Kr           = P3 + (P2 - P1),
  //   P1 = Wre@Kr, P2 = Wim@Ki, P3 = (Wre - Wim)@(Kr + Ki)
  v2f S1[8], S2[8], S3[8];
  {
    const float* Wb = W + (size_t)b * NR * MR * 2;
#pragma unroll
    for (int s = 0; s < 8; ++s) {
      const int l0 = 4 * s + 2 * half;   // K rows held by this lane half
      const float2 w0 = *(const float2*)(Wb + ((size_t)l0 * MR + l16) * 2);
      const float2 w1 = *(const float2*)(Wb + ((size_t)(l0 + 1) * MR + l16) * 2);
      S1[s].x = w0.x;        S1[s].y = w1.x;
      S2[s].x = w0.y;        S2[s].y = w1.y;
      S3[s].x = w0.x - w0.y; S3[s].y = w1.x - w1.y;
    }
  }

  // Phase-2 accumulators, Gauss form: Q0=Sum Fr@tRe, Q1=Sum Fi@tIm,
  // Q2=Sum (Fr+Fi)@(tRe+tIm). Combined once at the end.
  v8f Q[2][2][3];  // [jj][i-tile][product]
#pragma unroll
  for (int a0 = 0; a0 < 2; ++a0)
#pragma unroll
    for (int a1 = 0; a1 < 2; ++a1)
#pragma unroll
      for (int a2 = 0; a2 < 3; ++a2)
        Q[a0][a1][a2] = v8f{};

  const float* Fb = F + (size_t)b * NT * MT * 2;

#pragma unroll
  for (int kc = 0; kc < NCH; ++kc) {
    const int buf = kc & 1;
    const int k   = kc * KC + wave;   // each wave owns a distinct k

    // ---------------- Phase 1: issue all K2 loads up front ----------------
    float2 kv[16];
    {
      const float* Kb = K2 + ((size_t)k * NR * G2_ + g0 + l16) * 2;
#pragma unroll
      for (int s = 0; s < 8; ++s) {
        const int l0 = 4 * s + 2 * half;
        kv[2 * s + 0] = *(const float2*)(Kb + (size_t)l0 * G2_ * 2);
        kv[2 * s + 1] = *(const float2*)(Kb + (size_t)(l0 + 1) * G2_ * 2);
      }
    }

    // One prefetch covers the whole next-chunk slice for this wave's k:
    // lane -> row l (32 rows x 128B lines, 128B-aligned).
    if (kc + 1 < NCH) {
      __builtin_prefetch(
          K2 + (((size_t)(k + KC) * NR + lane) * G2_ + g0) * 2, 0, 3);
    }

    // 24 WMMAs over three independent accumulation chains
    v8f P1 = v8f{}, P2 = v8f{}, P3 = v8f{};
#pragma unroll
    for (int s = 0; s < 8; ++s) {
      v2f bRe, bIm, bS;
      bRe.x = kv[2 * s + 0].x; bRe.y = kv[2 * s + 1].x;
      bIm.x = kv[2 * s + 0].y; bIm.y = kv[2 * s + 1].y;
      bS.x  = bRe.x + bIm.x;   bS.y  = bRe.y + bIm.y;
      P1 = wmma_f32(S1[s], bRe, P1);
      P2 = wmma_f32(S2[s], bIm, P2);
      P3 = wmma_f32(S3[s], bS,  P3);
    }

    // combine + store tRe/tIm planes to the double buffer
#pragma unroll
    for (int v = 0; v < 8; ++v) {
      const int m = v + 8 * half;  // D-tile row = j
      const float tRe = P1[v] + P2[v];
      const float tIm = P3[v] + (P2[v] - P1[v]);
      tmpL[buf][((wave * 2 + 0) * 16 + m) * 16 + l16] = tRe;
      tmpL[buf][((wave * 2 + 1) * 16 + m) * 16 + l16] = tIm;
    }
    __syncthreads();

    // ---------------- Phase 2: accumulate over this 8-k chunk --------------
    v2f fr[2][2], fi[2][2], fs[2][2];  // [i-tile][k-step]
#pragma unroll
    for (int it = 0; it < 2; ++it)
#pragma unroll
      for (int t = 0; t < 2; ++t) {
        const int i  = it * 16 + l16;
        const int k0 = kc * KC + 4 * t + 2 * half;
        const float2 u0 = *(const float2*)(Fb + ((size_t)k0 * MT + i) * 2);
        const float2 u1 = *(const float2*)(Fb + ((size_t)(k0 + 1) * MT + i) * 2);
        fr[it][t].x = u0.x;        fr[it][t].y = u1.x;
        fi[it][t].x = u0.y;        fi[it][t].y = u1.y;
        fs[it][t].x = u0.x + u0.y; fs[it][t].y = u1.x + u1.y;
      }
#pragma unroll
    for (int jj = 0; jj < 2; ++jj) {
      const int j = wave * 2 + jj;
#pragma unroll
      for (int t = 0; t < 2; ++t) {
        const int kk0 = 4 * t + 2 * half;
        v2f bRe, bIm, bS;
        bRe.x = tmpL[buf][((kk0 * 2 + 0) * 16 + j) * 16 + l16];
        bRe.y = tmpL[buf][(((kk0 + 1) * 2 + 0) * 16 + j) * 16 + l16];
        bIm.x = tmpL[buf][((kk0 * 2 + 1) * 16 + j) * 16 + l16];
        bIm.y = tmpL[buf][(((kk0 + 1) * 2 + 1) * 16 + j) * 16 + l16];
        bS.x = bRe.x + bIm.x; bS.y = bRe.y + bIm.y;
#pragma unroll
        for (int it = 0; it < 2; ++it) {
          Q[jj][it][0] = wmma_f32(fr[it][t], bRe, Q[jj][it][0]);
          Q[jj][it][1] = wmma_f32(fi[it][t], bIm, Q[jj][it][1]);
          Q[jj][it][2] = wmma_f32(fs[it][t], bS,  Q[jj][it][2]);
        }
      }
    }
  }

  // ---- Combine Gauss products and store packed {Re, Im}, coalesced over g --
#pragma unroll
  for (int jj = 0; jj < 2; ++jj) {
    const int j = wave * 2 + jj;
#pragma unroll
    for (int it = 0; it < 2; ++it) {
#pragma unroll
      for (int v = 0; v < 8; ++v) {
        const int m = v + 8 * half;
        const int i = it * 16 + m;
        const float q0 = Q[jj][it][0][v];
        const float q1 = Q[jj][it][1][v];
        const float q2 = Q[jj][it][2][v];
        const size_t idx =
            (((size_t)b * (MT * MR) + (size_t)i * MR + j) * G2_ + g0 + l16) * 2;
        float2 o;
        o.x = q0 - q1;             // Re
        o.y = q2 - q0 - q1;        // Im
        *(float2*)(out + idx) = o;
      }
    }
  }
}

extern "C" void kernel_launch(void* const* d_in, const int* in_sizes, int n_in,
                              void* d_out, int out_size, void* d_ws, size_t ws_size,
                              hipStream_t stream) {
  const float* W  = (const float*)d_in[0];
  const float* F  = (const float*)d_in[1];
  const float* K2 = (const float*)d_in[2];
  float* out = (float*)d_out;
  dim3 grid(G2_ / GT, B_, 1);
  phi_kernel<<<grid, dim3(256), 0, stream>>>(W, F, K2, out);
}